// scaled_dot_product_attention_31971736551584
// MI455X (gfx1250) — compile-verified
//
#include <hip/hip_runtime.h>
#include <hip/hip_bf16.h>

typedef __attribute__((ext_vector_type(2))) float v2f;
typedef __attribute__((ext_vector_type(4))) float v4f;
typedef __attribute__((ext_vector_type(8))) float v8f;

#define B_SZ 16
#define LQ 2048
#define LK 2048
#define DH 64
#define DV 64
#define ROWSTRIDE 2052              // 2048 + 4 pad: 16B-aligned rows + LDS bank spread
#define NWAVES 8
#define NTHREADS 256
#define KT_PER_WAVE (LK / 16 / NWAVES)    // 16 score tiles per wave
#define KCH_PER_WAVE (LK / 4 / NWAVES)    // 64 K=4 chunks per wave in O phase

// LDS plan (dynamic, 136512 B; two WGs fit in the 320KB WGP LDS):
//   Sp   : float[16][ROWSTRIDE]  score / probability panel (131328 B)
//   oacc : float[1024]           64x16 O accumulator       (4096 B)
//   red  : float[256]            per-row 16-way reductions (1024 B)
//   invr : float[16]             per-row 1/rowsum          (64 B)

extern "C" __global__ __launch_bounds__(NTHREADS)
void attn_f32_wmma_kernel(const float* __restrict__ q,   // [B, LQ, DH]
                          const float* __restrict__ k,   // [B, DH, LK]  (pre-transposed)
                          const float* __restrict__ v,   // [B, DV, LK]
                          float* __restrict__ outp,      // [B, DV, LQ]
                          float* __restrict__ attnp)     // [B, LQ, LK]
{
    extern __shared__ float smem[];
    float* Sp   = smem;
    float* oacc = Sp + 16 * ROWSTRIDE;
    float* red  = oacc + 1024;
    float* invr = red + 256;

    const int tid  = threadIdx.x;
    const int lane = tid & 31;
    const int wave = tid >> 5;
    const int half = lane >> 4;       // wave32: lanes 0-15 / 16-31
    const int l16  = lane & 15;

    const int wg = blockIdx.x;
    const int b  = wg >> 7;           // 128 q-tiles per batch
    const int qt = wg & 127;

    const float* qb = q + ((size_t)b * LQ + (size_t)qt * 16) * DH;
    const float* kb = k + (size_t)b * DH * LK;
    const float* vb = v + (size_t)b * DV * LK;
    float* ob = outp  + (size_t)b * DV * LQ + (size_t)qt * 16;
    float* ab = attnp + ((size_t)b * LQ + (size_t)qt * 16) * LK;

    // zero the O accumulator tile
    for (int i = tid; i < 1024; i += NTHREADS) oacc[i] = 0.0f;

    // ---- Phase 0: load Q tile as 16 A-fragments (16x4 f32 each) ----
    // A layout: lane M = l16; VGPR0 = K(2*half), VGPR1 = K(2*half+1)  -> b64 load
    v2f qa[16];
#pragma unroll
    for (int c = 0; c < 16; ++c) {
        const float* p = qb + (size_t)l16 * DH + 4 * c + 2 * half;
        qa[c].x = p[0];
        qa[c].y = p[1];
    }

    // ---- Phase 1: S = Q * K  (this wave owns 16 of the 128 k-tiles) ----
    for (int t = 0; t < KT_PER_WAVE; ++t) {
        const int kt  = wave * KT_PER_WAVE + t;
        const int col = kt * 16 + l16;
        v8f acc = {};
#pragma unroll
        for (int c = 0; c < 16; ++c) {
            const int kd = 4 * c + 2 * half;
            v2f bf;                                    // B layout: lane N = l16
            bf.x = kb[(size_t)kd       * LK + col];
            bf.y = kb[(size_t)(kd + 1) * LK + col];
            acc = __builtin_amdgcn_wmma_f32_16x16x4_f32(
                      false, qa[c], false, bf, (short)0, acc, false, false);
        }
        // D layout: VGPR r -> row (r + 8*half), col = l16
#pragma unroll
        for (int r = 0; r < 8; ++r) {
            Sp[(r + 8 * half) * ROWSTRIDE + col] = acc[r];
        }
    }
    __syncthreads();

    // ---- Phase 2: row-wise softmax over the 16x2048 panel ----
    {
        const int row = tid >> 4;     // 16 rows x 16 threads each
        const int t16 = tid & 15;
        float* srow = Sp + row * ROWSTRIDE;

        float m = -3.402823466e38f;
        for (int c = t16; c < LK; c += 16) m = fmaxf(m, srow[c]);
        red[row * 16 + t16] = m;
        __syncthreads();
#pragma unroll
        for (int i = 0; i < 16; ++i) m = fmaxf(m, red[row * 16 + i]);

        float s = 0.0f;
        for (int c = t16; c < LK; c += 16) {
            float e = __expf(srow[c] - m);
            srow[c] = e;
            s += e;
        }
        __syncthreads();              // all reads of red (max) done
        red[row * 16 + t16] = s;
        __syncthreads();
        float tot = 0.0f;
#pragma unroll
        for (int i = 0; i < 16; ++i) tot += red[row * 16 + i];
        if (t16 == 0) invr[row] = 1.0f / tot;
    }
    __syncthreads();

    // ---- Phase 2b: normalize in LDS + fully-coalesced float4 attn store ----
    {
        const int nvec = 16 * (LK / 4);               // 8192 float4s
        for (int i = tid; i < nvec; i += NTHREADS) {
            const int row = i >> 9;                   // LK/4 = 512
            const int c4  = i & 511;
            const float inv = invr[row];
            float* sp4 = Sp + row * ROWSTRIDE + c4 * 4;   // 16B aligned (ROWSTRIDE%4==0)
            v4f val = *(v4f*)sp4;
            val *= inv;
            *(v4f*)sp4 = val;                         // keep normalized P for O phase
            *(v4f*)(ab + (size_t)row * LK + c4 * 4) = val;
        }
    }
    __syncthreads();

    // ---- Phase 3: O = V * P^T, each wave accumulates its k-slice ----
    v8f oa[4] = {{}, {}, {}, {}};
    const int kbase0 = wave * (LK / NWAVES);          // 256-wide slice
    for (int cc = 0; cc < KCH_PER_WAVE; ++cc) {
        const int kd = kbase0 + 4 * cc + 2 * half;
        v2f bf;                                       // B from LDS: lane q = l16, ds_load_b64
        const float* pp = Sp + l16 * ROWSTRIDE + kd;
        bf.x = pp[0];
        bf.y = pp[1];
#pragma unroll
        for (int vt = 0; vt < 4; ++vt) {
            const float* va = vb + (size_t)(vt * 16 + l16) * LK + kd;  // global b64
            v2f af;
            af.x = va[0];
            af.y = va[1];
            oa[vt] = __builtin_amdgcn_wmma_f32_16x16x4_f32(
                         false, af, false, bf, (short)0, oa[vt], false, false);
        }
    }
    // cross-wave reduction of partial O tiles via ds_add_f32
#pragma unroll
    for (int vt = 0; vt < 4; ++vt) {
#pragma unroll
        for (int r = 0; r < 8; ++r) {
            const int idx = vt * 256 + (r + 8 * half) * 16 + l16;  // == v*16 + q
            atomicAdd(&oacc[idx], oa[vt][r]);
        }
    }
    __syncthreads();

    // ---- Phase 4: write O tile (64 x 16) as float4 ----
    for (int i = tid; i < 256; i += NTHREADS) {
        const int idx  = i * 4;        // oacc linear index == v*16 + q
        const int vrow = idx >> 4;
        const int q4   = idx & 15;
        v4f val = *(v4f*)(oacc + idx);
        *(v4f*)(ob + (size_t)vrow * LQ + q4) = val;
    }
}

extern "C" void kernel_launch(void* const* d_in, const int* in_sizes, int n_in,
                              void* d_out, int out_size, void* d_ws, size_t ws_size,
                              hipStream_t stream) {
    // setup_inputs order: inputs (unused), query, key, value
    const float* q = (const float*)d_in[1];
    const float* k = (const float*)d_in[2];
    const float* v = (const float*)d_in[3];
    float* outp  = (float*)d_out;                          // [B, DV, LQ] first
    float* attnp = outp + (size_t)B_SZ * DV * LQ;          // then [B, LQ, LK]

    const size_t shmem = (size_t)(16 * ROWSTRIDE + 1024 + 256 + 16) * sizeof(float);
    dim3 grid(B_SZ * (LQ / 16));   // 2048 workgroups
    attn_f32_wmma_kernel<<<grid, NTHREADS, shmem, stream>>>(q, k, v, outp, attnp);
}